// GCNLayer_3255585210595
// MI455X (gfx1250) — compile-verified
//
#include <hip/hip_runtime.h>
#include <hip/hip_bf16.h>

typedef __attribute__((ext_vector_type(2))) float v2f;
typedef __attribute__((ext_vector_type(4))) float v4f;
typedef __attribute__((ext_vector_type(8))) float v8f;

#define DIM 128
#define BN_EPS 1e-5f

// ---------------------------------------------------------------------------
// 1) Edge scatter: agg[dst] += feature[src], deg[dst] += 1
//    One thread per (edge, dim). 128 consecutive threads cover one edge's
//    512B row -> coalesced gather; atomics land in L2 (agg is 51MB < 192MB L2).
// ---------------------------------------------------------------------------
__global__ void __launch_bounds__(256) scatter_kernel(
    const float* __restrict__ feat, const int* __restrict__ src,
    const int* __restrict__ dst, float* __restrict__ agg,
    float* __restrict__ deg, int E) {
  long idx = (long)blockIdx.x * 256 + threadIdx.x;
  int e = (int)(idx >> 7);
  if (e >= E) return;
  int d = (int)(idx & (DIM - 1));
  int s = src[e];
  int t = dst[e];
  unsafeAtomicAdd(&agg[(size_t)t * DIM + d], feat[(size_t)s * DIM + d]);
  if (d == 0) unsafeAtomicAdd(&deg[t], 1.0f);
}

// ---------------------------------------------------------------------------
// 2) invdeg[i] = 1 / max(deg[i], 1)
// ---------------------------------------------------------------------------
__global__ void __launch_bounds__(256) invdeg_kernel(
    const float* __restrict__ deg, float* __restrict__ invdeg, int N) {
  int i = blockIdx.x * 256 + threadIdx.x;
  if (i < N) invdeg[i] = 1.0f / fmaxf(deg[i], 1.0f);
}

// ---------------------------------------------------------------------------
// 3) GEMM: hlin = (agg * invdeg_row) @ W.T + b  via V_WMMA_F32_16X16X4_F32.
//    Block = 256 threads = 8 waves; block owns 16 rows (m-tile), wave w owns
//    output columns [16w, 16w+16). K=128 -> 32 chained WMMAs per wave.
//    A lane layout: lane l, vgpr v -> A[l%16][2*(l/16)+v]  (float2 load)
//    B[k][n] = W[n][k]            -> same float2 pattern from W row n.
// ---------------------------------------------------------------------------
__global__ void __launch_bounds__(256) gemm_wmma_kernel(
    const float* __restrict__ agg, const float* __restrict__ invdeg,
    const float* __restrict__ W, const float* __restrict__ bias,
    float* __restrict__ hlin, int N) {
  int tid = threadIdx.x;
  int wv = tid >> 5;        // wave id 0..7 -> n-tile
  int l  = tid & 31;        // lane
  int hi = l >> 4;          // 0/1
  int lm = l & 15;          // 0..15
  int m_base = blockIdx.x * 16;
  int n_base = wv * 16;

  int row_a = m_base + lm;
  if (row_a >= N) row_a = N - 1;           // clamp (stores are predicated)
  float inv = invdeg[row_a];

  const float* ap = agg + (size_t)row_a * DIM + 2 * hi;
  const float* bp = W + (size_t)(n_base + lm) * DIM + 2 * hi;

  v8f c = {};
#pragma unroll
  for (int k = 0; k < DIM; k += 4) {
    v2f a = *(const v2f*)(ap + k);
    a = a * inv;                            // fold mean-divide into A
    v2f bb = *(const v2f*)(bp + k);
    c = __builtin_amdgcn_wmma_f32_16x16x4_f32(
        /*neg_a=*/false, a, /*neg_b=*/false, bb,
        /*c_mod=*/(short)0, c, /*reuse_a=*/false, /*reuse_b=*/false);
  }

  float bv = bias[n_base + lm];
#pragma unroll
  for (int v = 0; v < 8; ++v) {
    int row = m_base + v + 8 * hi;          // C layout: M = v + 8*(l/16)
    if (row < N) hlin[(size_t)row * DIM + n_base + lm] = c[v] + bv;
  }
}

// ---------------------------------------------------------------------------
// 4) Column stats: per-column sum and sum-of-squares over rows (batch dim).
//    256 threads: col = t%128, two row-lanes; register partials, then one
//    fp32 atomic per thread per output. Reads are fully coalesced row-major.
// ---------------------------------------------------------------------------
__global__ void __launch_bounds__(256) colstats_kernel(
    const float* __restrict__ x, long nrows, int rows_per_block,
    float* __restrict__ sum_out, float* __restrict__ sq_out) {
  int col = threadIdx.x & (DIM - 1);
  int half = threadIdx.x >> 7;             // 0 or 1
  long rbase = (long)blockIdx.x * rows_per_block;
  long rend = rbase + rows_per_block;
  if (rend > nrows) rend = nrows;
  float s = 0.0f, q = 0.0f;
  for (long r = rbase + half; r < rend; r += 2) {
    float v = x[r * DIM + col];
    s += v;
    q += v * v;
  }
  unsafeAtomicAdd(&sum_out[col], s);
  unsafeAtomicAdd(&sq_out[col], q);
}

// ---------------------------------------------------------------------------
// 5) Turn stats into per-column affine: scale = gamma*rsqrt(var+eps),
//    shift = beta - mean*scale.  t<128 -> h, t>=128 -> e. One tiny block.
// ---------------------------------------------------------------------------
__global__ void prep_kernel(const float* __restrict__ stats,
                            const float* __restrict__ gamma_h,
                            const float* __restrict__ beta_h,
                            const float* __restrict__ gamma_e,
                            const float* __restrict__ beta_e,
                            float Ncnt, float Ecnt,
                            float* __restrict__ scaleshift) {
  int t = threadIdx.x;
  int col = t & (DIM - 1);
  int which = t >> 7;                      // 0 = h, 1 = e
  float cnt = which ? Ecnt : Ncnt;
  float sum = stats[which * 256 + col];
  float sq  = stats[which * 256 + DIM + col];
  float mean = sum / cnt;
  float var = fmaxf(sq / cnt - mean * mean, 0.0f);
  float g = which ? gamma_e[col] : gamma_h[col];
  float be = which ? beta_e[col] : beta_h[col];
  float sc = g * rsqrtf(var + BN_EPS);
  scaleshift[which * 256 + col] = sc;
  scaleshift[which * 256 + DIM + col] = be - mean * sc;
}

// ---------------------------------------------------------------------------
// 6) Fused BN + ReLU + residual, float4-vectorized:
//    out = resid + max(x*scale[col] + shift[col], 0)
// ---------------------------------------------------------------------------
__global__ void __launch_bounds__(256) finalize_kernel(
    const float* __restrict__ x, const float* __restrict__ resid,
    const float* __restrict__ scale, const float* __restrict__ shift,
    float* __restrict__ out, long total4) {
  long i = (long)blockIdx.x * 256 + threadIdx.x;
  if (i >= total4) return;
  int col4 = (int)((i & (DIM / 4 - 1)) << 2);   // (i*4) % 128
  v4f xv = *(const v4f*)(x + i * 4);
  v4f rv = *(const v4f*)(resid + i * 4);
  v4f sc = *(const v4f*)(scale + col4);
  v4f sh = *(const v4f*)(shift + col4);
  v4f o;
#pragma unroll
  for (int j = 0; j < 4; ++j)
    o[j] = rv[j] + fmaxf(xv[j] * sc[j] + sh[j], 0.0f);
  *(v4f*)(out + i * 4) = o;
}

// ---------------------------------------------------------------------------
extern "C" void kernel_launch(void* const* d_in, const int* in_sizes, int n_in,
                              void* d_out, int out_size, void* d_ws, size_t ws_size,
                              hipStream_t stream) {
  const float* feature = (const float*)d_in[0];
  const float* e_in    = (const float*)d_in[1];
  const int*   src     = (const int*)d_in[2];
  const int*   dst     = (const int*)d_in[3];
  const float* W       = (const float*)d_in[4];
  const float* b       = (const float*)d_in[5];
  const float* gamma_h = (const float*)d_in[6];
  const float* beta_h  = (const float*)d_in[7];
  const float* gamma_e = (const float*)d_in[8];
  const float* beta_e  = (const float*)d_in[9];

  const long N = in_sizes[0] / DIM;   // 100000
  const long E = in_sizes[2];         // 1600000

  float* h_out = (float*)d_out;             // [N, 128]
  float* e_out = h_out + (size_t)N * DIM;   // [E, 128]

  // workspace layout
  float* agg        = (float*)d_ws;                 // N*DIM  (zeroed)
  float* deg        = agg + (size_t)N * DIM;        // N      (zeroed)
  float* stats      = deg + N;                      // 512    (zeroed): h_sum,h_sq,e_sum,e_sq
  float* invdeg     = stats + 512;                  // N      (fully written)
  float* scaleshift = invdeg + N;                   // 512    (fully written)

  size_t zero_floats = (size_t)N * DIM + (size_t)N + 512;
  hipMemsetAsync(agg, 0, zero_floats * sizeof(float), stream);

  // 1) edge scatter-add (mean aggregation numerator + degree)
  long totalED = E * DIM;
  scatter_kernel<<<(totalED + 255) / 256, 256, 0, stream>>>(feature, src, dst,
                                                            agg, deg, (int)E);
  // 2) reciprocal degree
  invdeg_kernel<<<(N + 255) / 256, 256, 0, stream>>>(deg, invdeg, (int)N);

  // 3) WMMA GEMM: hlin = (agg/deg) @ W.T + b   -> written to h region of d_out
  gemm_wmma_kernel<<<(N + 15) / 16, 256, 0, stream>>>(agg, invdeg, W, b,
                                                      h_out, (int)N);

  // 4) batch-dim column stats for h and e
  colstats_kernel<<<(N + 255) / 256, 256, 0, stream>>>(h_out, N, 256,
                                                       stats + 0, stats + DIM);
  colstats_kernel<<<(E + 2047) / 2048, 256, 0, stream>>>(e_in, E, 2048,
                                                         stats + 256, stats + 256 + DIM);
  // 5) per-column affine
  prep_kernel<<<1, 256, 0, stream>>>(stats, gamma_h, beta_h, gamma_e, beta_e,
                                     (float)N, (float)E, scaleshift);

  // 6) fused BN + ReLU + residual
  long h4 = (long)N * DIM / 4;
  finalize_kernel<<<(h4 + 255) / 256, 256, 0, stream>>>(
      h_out, feature, scaleshift, scaleshift + DIM, h_out, h4);
  long e4 = (long)E * DIM / 4;
  finalize_kernel<<<(e4 + 255) / 256, 256, 0, stream>>>(
      e_in, e_in, scaleshift + 256, scaleshift + 256 + DIM, e_out, e4);
}